// MultiHeadAttention_5085241278813
// MI455X (gfx1250) — compile-verified
//
#include <hip/hip_runtime.h>

typedef float v2f __attribute__((ext_vector_type(2)));
typedef float v8f __attribute__((ext_vector_type(8)));

// ---------------------------------------------------------------------------
// Generic f32 GEMM: C[M,N] = A[M,K] * B[K,N], via V_WMMA_F32_16X16X4_F32.
// Block = 256 threads (8 waves). Block tile = 128(M) x 64(N), K-panel = 32.
// Wave w owns rows [w*16, w*16+16) x all 64 block columns (4 accumulators).
// A fragment (16x4 f32): lane L row = L&15; VGPR0/1 hold K = {kk,kk+1} for
// lanes 0-15 and K = {kk+2,kk+3} for lanes 16-31  -> one float2 LDS read,
// reused across the 4 N-subtiles. B panel stored transposed in LDS so each
// B fragment is also one contiguous float2 read.
// ---------------------------------------------------------------------------
#define TBK 32
#define TN  64

__global__ __launch_bounds__(256) void gemm_f32_wmma(
    const float* __restrict__ A, const float* __restrict__ B,
    float* __restrict__ C, int M, int N, int K)
{
    __shared__ float As[128 * TBK];   // row-major  As[row*TBK + k]   (16 KB)
    __shared__ float Bs[TN  * TBK];   // transposed Bs[col*TBK + k]   ( 8 KB)

    const int tid  = threadIdx.x;
    const int lane = tid & 31;
    const int wave = tid >> 5;
    const int mB   = blockIdx.x * 128;
    const int nB   = blockIdx.y * TN;
    const int col  = lane & 15;
    const int koff = (lane >> 4) << 1;        // 0 for lanes 0-15, 2 for 16-31

    const v8f vzero = {0.f,0.f,0.f,0.f,0.f,0.f,0.f,0.f};
    v8f acc[4];
    #pragma unroll
    for (int c = 0; c < 4; ++c) acc[c] = vzero;
    const int aRow = wave * 16 + col;

    for (int k0 = 0; k0 < K; k0 += TBK) {
        // stage A panel: 128x32 f32 = 1024 float4, 4 per thread
        #pragma unroll
        for (int i = 0; i < 4; ++i) {
            int f   = tid + i * 256;          // float4 granule 0..1023
            int row = f >> 3;                 // 8 float4 per row
            int c4  = f & 7;
            *(float4*)(As + row * TBK + c4 * 4) =
                *(const float4*)(A + (size_t)(mB + row) * K + k0 + c4 * 4);
            if (k0 + TBK < K)                 // CDNA5 global_prefetch_b8
                __builtin_prefetch(A + (size_t)(mB + row) * K + k0 + TBK + c4 * 4, 0, 1);
        }
        // stage B panel transposed: 32x64 f32 = 2048 floats, 8 per thread
        #pragma unroll
        for (int i = 0; i < 8; ++i) {
            int e  = tid + i * 256;           // 0..2047
            int kr = e >> 6;                  // 0..31
            int n  = e & 63;
            Bs[n * TBK + kr] = B[(size_t)(k0 + kr) * N + nB + n];
        }
        __syncthreads();

        #pragma unroll
        for (int kk = 0; kk < TBK; kk += 4) {
            v2f a = *(const v2f*)(As + aRow * TBK + kk + koff);
            #pragma unroll
            for (int c = 0; c < 4; ++c) {
                v2f b = *(const v2f*)(Bs + (c * 16 + col) * TBK + kk + koff);
                acc[c] = __builtin_amdgcn_wmma_f32_16x16x4_f32(
                             false, a, false, b, (short)0, acc[c], false, false);
            }
        }
        __syncthreads();
    }

    // C/D layout: VGPR j, lanes 0-15 -> M=j, lanes 16-31 -> M=j+8; N = lane&15
    const int mBase = mB + wave * 16 + 8 * (lane >> 4);
    #pragma unroll
    for (int c = 0; c < 4; ++c) {
        const int n = nB + c * 16 + col;
        #pragma unroll
        for (int j = 0; j < 8; ++j)
            C[(size_t)(mBase + j) * N + n] = acc[c][j];
    }
}

// ---------------------------------------------------------------------------
// Zero-centered RMSNorm + RoPE. One wave per (token, head); lane owns 4 elems.
// proj: [B*T, NH*128]  ->  out: [B, NH, T, 128]
// ---------------------------------------------------------------------------
__global__ __launch_bounds__(256) void norm_rope(
    const float* __restrict__ proj, const float* __restrict__ scale,
    const float* __restrict__ cs, const float* __restrict__ sn,
    float* __restrict__ out, int NH, int T)
{
    const int gw   = (blockIdx.x * 256 + threadIdx.x) >> 5;
    const int lane = threadIdx.x & 31;
    const int h  = gw % NH;
    const int bt = gw / NH;
    const int t  = bt % T;
    const int b  = bt / T;

    const float* x = proj + (size_t)bt * (NH * 128) + h * 128;
    float4 xv = *(const float4*)(x + lane * 4);

    float ss = xv.x*xv.x + xv.y*xv.y + xv.z*xv.z + xv.w*xv.w;
    ss += __shfl_xor(ss, 1, 32);
    ss += __shfl_xor(ss, 2, 32);
    ss += __shfl_xor(ss, 4, 32);
    ss += __shfl_xor(ss, 8, 32);
    ss += __shfl_xor(ss, 16, 32);
    const float inv = rsqrtf(ss * (1.0f / 128.0f) + 1e-6f);

    float4 scv = *(const float4*)(scale + lane * 4);
    float r0 = (1.0f + scv.x) * xv.x * inv;
    float r1 = (1.0f + scv.y) * xv.y * inv;
    float r2 = (1.0f + scv.z) * xv.z * inv;
    float r3 = (1.0f + scv.w) * xv.w * inv;

    // RoPE: lanes 0-15 hold x1 (d<64), lanes 16-31 hold x2 (d>=64)
    const int dh = (lane & 15) * 4;
    float4 c = *(const float4*)(cs + (size_t)t * 64 + dh);
    float4 s = *(const float4*)(sn + (size_t)t * 64 + dh);
    float p0 = __shfl_xor(r0, 16, 32);
    float p1 = __shfl_xor(r1, 16, 32);
    float p2 = __shfl_xor(r2, 16, 32);
    float p3 = __shfl_xor(r3, 16, 32);

    float4 ov;
    if (lane < 16) {
        ov.x = r0*c.x - p0*s.x; ov.y = r1*c.y - p1*s.y;
        ov.z = r2*c.z - p2*s.z; ov.w = r3*c.w - p3*s.w;
    } else {
        ov.x = r0*c.x + p0*s.x; ov.y = r1*c.y + p1*s.y;
        ov.z = r2*c.z + p2*s.z; ov.w = r3*c.w + p3*s.w;
    }
    *(float4*)(out + (((size_t)b * NH + h) * T + t) * 128 + lane * 4) = ov;
}

// ---------------------------------------------------------------------------
// [B*T, NH*128] -> [B, NH, T, 128] copy (for V).
// ---------------------------------------------------------------------------
__global__ __launch_bounds__(256) void reshape_hs(
    const float* __restrict__ in, float* __restrict__ out, int NH, int T)
{
    int idx  = blockIdx.x * 256 + threadIdx.x;   // float4 index
    int d4   = idx & 31;
    int rest = idx >> 5;
    int h    = rest % NH;
    int bt   = rest / NH;
    int t = bt % T, b = bt / T;
    float4 v = *(const float4*)(in + (size_t)bt * (NH * 128) + h * 128 + d4 * 4);
    *(float4*)(out + (((size_t)b * NH + h) * T + t) * 128 + d4 * 4) = v;
}

// ---------------------------------------------------------------------------
// Flash attention, causal, GQA (H=16, KVH=4, HD=128). One wave per 16-row
// Q tile; block = 64 threads = 2 independent waves. Online softmax; QK^T and
// P*V through V_WMMA_F32_16X16X4_F32. Row stats (max/sum) live in registers,
// replicated across the 16-lane group that owns each C-fragment row.
// Output written as [B*T, 2048] with column h*128+d (ready for out-proj GEMM).
// ---------------------------------------------------------------------------
__global__ __launch_bounds__(64) void attn_flash(
    const float* __restrict__ q,   // [B,16,T,128]
    const float* __restrict__ k,   // [B,4,T,128]
    const float* __restrict__ v,   // [B,4,T,128]
    float* __restrict__ o,         // [B*T, 2048]
    int T)
{
    constexpr int H = 16, KVH = 4, HD = 128;
    __shared__ float smem[2 * 6400];             // per-wave: Q,K,V 16x128 + P 16x16
    const int lane = threadIdx.x & 31;
    const int wave = threadIdx.x >> 5;
    float* Qs = smem + wave * 6400;
    float* Ks = Qs + 2048;
    float* Vs = Ks + 2048;
    float* Ps = Vs + 2048;

    const int nQT = T / 16;
    const int gw  = blockIdx.x * 2 + wave;
    const int qt  = gw % nQT;
    const int h   = (gw / nQT) % H;
    const int b   = gw / (nQT * H);
    const int kvh = h >> 2;                      // rep = H/KVH = 4

    const int qbase = qt * 16;
    const float* qp = q + (((size_t)b * H   + h  ) * T + qbase) * HD;
    const float* kp = k + (((size_t)b * KVH + kvh) * T) * HD;
    const float* vp = v + (((size_t)b * KVH + kvh) * T) * HD;

    // stage Q tile (16x128) in LDS
    #pragma unroll
    for (int i = 0; i < 16; ++i)
        *(float4*)(Qs + i * HD + lane * 4) =
            *(const float4*)(qp + (size_t)i * HD + lane * 4);

    const int col  = lane & 15;
    const int half = lane >> 4;
    const int koff = half << 1;

    const v8f vzero = {0.f,0.f,0.f,0.f,0.f,0.f,0.f,0.f};
    v8f acc[8];
    #pragma unroll
    for (int c = 0; c < 8; ++c) acc[c] = vzero;
    float rowm[8], rowl[8];
    #pragma unroll
    for (int j = 0; j < 8; ++j) { rowm[j] = -1e30f; rowl[j] = 0.f; }

    const float sc = 0.08838834764831845f;       // 1/sqrt(128)

    for (int kt = 0; kt <= qt; ++kt) {
        const int kb = kt * 16;
        // stage K,V tiles (16x128 each)
        #pragma unroll
        for (int i = 0; i < 16; ++i) {
            *(float4*)(Ks + i * HD + lane * 4) =
                *(const float4*)(kp + (size_t)(kb + i) * HD + lane * 4);
            *(float4*)(Vs + i * HD + lane * 4) =
                *(const float4*)(vp + (size_t)(kb + i) * HD + lane * 4);
        }
        asm volatile("s_wait_dscnt 0x0" ::: "memory");

        // S = Q * K^T  (16x16, K-dim = 128)
        v8f s = vzero;
        #pragma unroll
        for (int kk = 0; kk < HD; kk += 4) {
            v2f a  = *(const v2f*)(Qs + col * HD + kk + koff);
            v2f bb = *(const v2f*)(Ks + col * HD + kk + koff);
            s = __builtin_amdgcn_wmma_f32_16x16x4_f32(
                    false, a, false, bb, (short)0, s, false, false);
        }

        // online softmax update
        const bool diag = (kt == qt);
        float fac[8];
        #pragma unroll
        for (int j = 0; j < 8; ++j) {
            float sv = s[j] * sc;
            const int m = j + 8 * half;          // global row of this element
            if (diag && (kb + col > qbase + m)) sv = -1e30f;
            float tm = sv;                       // row max over 16-lane group
            tm = fmaxf(tm, __shfl_xor(tm, 1, 32));
            tm = fmaxf(tm, __shfl_xor(tm, 2, 32));
            tm = fmaxf(tm, __shfl_xor(tm, 4, 32));
            tm = fmaxf(tm, __shfl_xor(tm, 8, 32));
            const float newm = fmaxf(rowm[j], tm);
            fac[j] = __expf(rowm[j] - newm);
            const float pv = __expf(sv - newm);
            float ts = pv;                       // row sum over 16-lane group
            ts += __shfl_xor(ts, 1, 32);
            ts += __shfl_xor(ts, 2, 32);
            ts += __shfl_xor(ts, 4, 32);
            ts += __shfl_xor(ts, 8, 32);
            rowl[j] = rowl[j] * fac[j] + ts;
            rowm[j] = newm;
            Ps[(j + 8 * half) * 16 + col] = pv;  // P tile to LDS (A-matrix feed)
        }
        asm volatile("s_wait_dscnt 0x0" ::: "memory");

        // rescale accumulator rows
        #pragma unroll
        for (int c = 0; c < 8; ++c)
            #pragma unroll
            for (int j = 0; j < 8; ++j) acc[c][j] *= fac[j];

        // O += P * V   (16x128, K-dim = 16, 8 column chunks)
        #pragma unroll
        for (int c = 0; c < 8; ++c) {
            const int nb = c * 16;
            #pragma unroll
            for (int kk = 0; kk < 16; kk += 4) {
                v2f a = *(const v2f*)(Ps + col * 16 + kk + koff);
                v2f bb;
                bb.x = Vs[(kk + koff    ) * HD + nb + col];
                bb.y = Vs[(kk + koff + 1) * HD + nb + col];
                acc[c] = __builtin_amdgcn_wmma_f32_16x16x4_f32(
                             false, a, false, bb, (short)0, acc[c], false, false);
            }
        }
    }

    // epilogue: divide by row sums, write [B*T, 2048] at column h*128+d
    float inv[8];
    #pragma unroll
    for (int j = 0; j < 8; ++j) inv[j] = 1.0f / rowl[j];
    #pragma unroll
    for (int c = 0; c < 8; ++c)
        #pragma unroll
        for (int j = 0; j < 8; ++j) {
            const int m = j + 8 * half;
            o[((size_t)(b * T + qbase + m)) * 2048 + h * 128 + c * 16 + col] =
                acc[c][j] * inv[j];
        }
}

// ---------------------------------------------------------------------------
extern "C" void kernel_launch(void* const* d_in, const int* in_sizes, int n_in,
                              void* d_out, int out_size, void* d_ws, size_t ws_size,
                              hipStream_t stream)
{
    (void)in_sizes; (void)n_in; (void)out_size; (void)ws_size;

    const float* q_input  = (const float*)d_in[0];
    const float* kv_input = (const float*)d_in[1];
    /* d_in[2] = mask: causal, computed analytically */
    const float* cosT     = (const float*)d_in[3];
    const float* sinT     = (const float*)d_in[4];
    const float* Wq       = (const float*)d_in[5];
    const float* Wk       = (const float*)d_in[6];
    const float* Wv       = (const float*)d_in[7];
    const float* q_scale  = (const float*)d_in[8];
    const float* k_scale  = (const float*)d_in[9];
    const float* Wo       = (const float*)d_in[10];
    float* out = (float*)d_out;

    constexpr int B = 2, T = 2048, D = 2048, H = 16, KVH = 4, HD = 128;
    constexpr int BT = B * T;
    constexpr size_t KVD = (size_t)KVH * HD;     // 512

    // workspace layout (~96 MiB of f32); qp is reused as the attention output
    float* ws = (float*)d_ws;
    float* qp = ws;                              // BT*D
    float* kp = qp + (size_t)BT * D;             // BT*512
    float* vp = kp + (size_t)BT * KVD;           // BT*512
    float* qr = vp + (size_t)BT * KVD;           // BT*D   [B,H,T,HD]
    float* kr = qr + (size_t)BT * D;             // BT*512 [B,KVH,T,HD]
    float* vr = kr + (size_t)BT * KVD;           // BT*512 [B,KVH,T,HD]

    const dim3 blk(256);

    // projections (WMMA GEMMs), 128x64 block tiles
    gemm_f32_wmma<<<dim3(BT / 128, D / TN),        blk, 0, stream>>>(q_input,  Wq, qp, BT, D,        D);
    gemm_f32_wmma<<<dim3(BT / 128, (int)KVD / TN), blk, 0, stream>>>(kv_input, Wk, kp, BT, (int)KVD, D);
    gemm_f32_wmma<<<dim3(BT / 128, (int)KVD / TN), blk, 0, stream>>>(kv_input, Wv, vp, BT, (int)KVD, D);

    // RMSNorm + RoPE (+ head-major relayout)
    norm_rope<<<(BT * H)   / 8, blk, 0, stream>>>(qp, q_scale, cosT, sinT, qr, H,   T);
    norm_rope<<<(BT * KVH) / 8, blk, 0, stream>>>(kp, k_scale, cosT, sinT, kr, KVH, T);
    reshape_hs<<<(BT * (int)KVD / 4) / 256, blk, 0, stream>>>(vp, vr, KVH, T);

    // flash attention (writes [BT, D] into qp)
    attn_flash<<<(B * H * (T / 16)) / 2, dim3(64), 0, stream>>>(qr, kr, vr, qp, T);

    // output projection
    gemm_f32_wmma<<<dim3(BT / 128, D / TN), blk, 0, stream>>>(qp, Wo, out, BT, D, D);
}